// TFNFusion_89550068122162
// MI455X (gfx1250) — compile-verified
//
#include <hip/hip_runtime.h>
#include <hip/hip_bf16.h>

typedef float v2f __attribute__((ext_vector_type(2)));
typedef float v8f __attribute__((ext_vector_type(8)));

#define HH   128
#define HH1  129
#define SLAB (HH1 * 64)   // floats per (i,j) slab of W1: 129 rows x 64 cols

// ---------------- prep: transpose v -> vT[j][m], zero the accumulator ----------------
__global__ void tfn_prep(const float* __restrict__ v,
                         float* __restrict__ vT,
                         float* __restrict__ Cacc) {
  int idx = blockIdx.x * blockDim.x + threadIdx.x;
  if (idx < 64 * 64) Cacc[idx] = 0.0f;
  for (int p = idx; p < HH1 * 64; p += gridDim.x * blockDim.x) {
    int j = p >> 6, m = p & 63;
    vT[p] = (j == 0) ? 1.0f : v[m * HH + (j - 1)];
  }
}

// ---------------- main: C[64,64] += fusion @ W1 via f32 WMMA, split-K over (i, j-chunk) ----------------
__global__ __launch_bounds__(512) void tfn_main(const float* __restrict__ a,
                                                const float* __restrict__ t,
                                                const float* __restrict__ W1,
                                                const float* __restrict__ vT,
                                                float* __restrict__ Cacc) {
  const int i  = blockIdx.x >> 2;          // 0..128 (a_h index)
  const int jq = blockIdx.x & 3;           // j-chunk
  const int j0 = (jq * HH1) >> 2;
  const int j1 = ((jq + 1) * HH1) >> 2;    // [0,32),[32,64),[64,96),[96,129)

  const int w     = threadIdx.x >> 5;      // wave 0..15
  const int lane  = threadIdx.x & 31;
  const int mtile = (w & 3) * 16;          // batch-tile base
  const int ntile = (w >> 2) * 16;         // out-feature strip base
  const int l15   = lane & 15;
  const int kh    = lane >> 4;             // K-half for A/B fragments

  // ---- resident A fragments: A[m,k] = t_h[m,k], constant across all slabs ----
  // 16x4 f32 A layout: lanes 0-15 M, V0:K=2*kh, V1:K=2*kh+1
  const int mA = mtile + l15;
  v2f At[33];
#pragma unroll
  for (int s = 0; s < 32; ++s) {
    const int k0 = 4 * s + 2 * kh;
    At[s].x = (k0 == 0) ? 1.0f : t[mA * HH + k0 - 1];  // t_h[mA, k0]
    At[s].y = t[mA * HH + k0];                          // t_h[mA, k0+1]
  }
  At[32].x = (kh == 0) ? t[mA * HH + 127] : 0.0f;       // tail k=128, lanes16-31 zeroed
  At[32].y = 0.0f;

  // ---- a_h column i in C/D layout: row m = mtile + r + 8*kh ----
  float av[8];
#pragma unroll
  for (int r = 0; r < 8; ++r) {
    const int m = mtile + r + 8 * kh;
    av[r] = (i == 0) ? 1.0f : a[m * HH + i - 1];
  }

  const int nb = ntile + l15;              // W1 column for B fragment
  v8f acc = {};

  const float* wrow = W1 + (size_t)i * HH1 * SLAB;
  for (int j = j0; j < j1; ++j) {
    const float* wp = wrow + (size_t)j * SLAB;
    const float* pw = wp + (2 * kh) * 64 + nb;   // B: V0 row 4s+2kh, V1 row 4s+2kh+1

    // per-slab scale s[m] = a_h[m,i] * v_h[m,j] in C/D row layout
    float sv[8];
#pragma unroll
    for (int r = 0; r < 8; ++r) {
      const int m = mtile + r + 8 * kh;
      sv[r] = av[r] * vT[j * 64 + m];
    }

    v8f P = {};
#pragma unroll
    for (int s = 0; s < 32; ++s) {
      v2f B;
      B.x = pw[(4 * s) * 64];
      B.y = pw[(4 * s + 1) * 64];
      P = __builtin_amdgcn_wmma_f32_16x16x4_f32(false, At[s], false, B,
                                                (short)0, P, false, false);
    }
    {   // K remainder (k=128): only lanes 0-15 of V0 carry data; matching A lanes are 0
      v2f B;
      B.x = wp[128 * 64 + nb];
      B.y = 0.0f;
      P = __builtin_amdgcn_wmma_f32_16x16x4_f32(false, At[32], false, B,
                                                (short)0, P, false, false);
    }

#pragma unroll
    for (int r = 0; r < 8; ++r) acc[r] = fmaf(sv[r], P[r], acc[r]);

    // prefetch next slab's W1 (speculative; drops silently past end)
    __builtin_prefetch(wp + SLAB + lane * 64, 0, 1);
  }

  // accumulate partial 64x64 result
#pragma unroll
  for (int r = 0; r < 8; ++r) {
    const int m = mtile + r + 8 * kh;
    unsafeAtomicAdd(&Cacc[m * 64 + nb], acc[r]);
  }
}

// ---------------- tail: bias+ReLU, y1@W2+b2 -> ReLU -> @W3+b3 ----------------
__global__ void tfn_tail(const float* __restrict__ Cacc,
                         const float* __restrict__ b1,
                         const float* __restrict__ W2,
                         const float* __restrict__ b2,
                         const float* __restrict__ W3,
                         const float* __restrict__ b3,
                         float* __restrict__ out) {
  const int b = threadIdx.x;
  if (b >= 64) return;
  float y1[64];
#pragma unroll
  for (int o = 0; o < 64; ++o) y1[o] = fmaxf(Cacc[b * 64 + o] + b1[o], 0.0f);
  float outv = b3[0];
  for (int o2 = 0; o2 < 32; ++o2) {
    float s = b2[o2];
#pragma unroll
    for (int o = 0; o < 64; ++o) s = fmaf(y1[o], W2[o * 32 + o2], s);
    outv = fmaf(fmaxf(s, 0.0f), W3[o2], outv);
  }
  out[b] = outv;
}

extern "C" void kernel_launch(void* const* d_in, const int* in_sizes, int n_in,
                              void* d_out, int out_size, void* d_ws, size_t ws_size,
                              hipStream_t stream) {
  const float* v  = (const float*)d_in[0];
  const float* a  = (const float*)d_in[1];
  const float* t  = (const float*)d_in[2];
  const float* W1 = (const float*)d_in[3];
  const float* b1 = (const float*)d_in[4];
  const float* W2 = (const float*)d_in[5];
  const float* b2 = (const float*)d_in[6];
  const float* W3 = (const float*)d_in[7];
  const float* b3 = (const float*)d_in[8];
  float* out  = (float*)d_out;

  float* vT   = (float*)d_ws;       // 129*64 floats
  float* Cacc = vT + HH1 * 64;      // 64*64 floats

  tfn_prep<<<33, 256, 0, stream>>>(v, vT, Cacc);
  tfn_main<<<129 * 4, 512, 0, stream>>>(a, t, W1, vT, Cacc);
  tfn_tail<<<1, 64, 0, stream>>>(Cacc, b1, W2, b2, W3, b3, out);
}